// NA_LDA_10909216931906
// MI455X (gfx1250) — compile-verified
//
#include <hip/hip_runtime.h>
#include <hip/hip_bf16.h>
#include <math.h>

typedef __attribute__((ext_vector_type(2))) float v2f;
typedef __attribute__((ext_vector_type(8))) float v8f;

#define Bn 16
#define Cn 256
#define Hn 64
#define Wn 64
#define HALF 128   // C/2
#define PLANE (Hn * Wn)   // 4096

// ---------------- workspace layout (floats) ----------------
#define WS_WW     0
#define WS_WH     256
#define WS_GHPART 512
#define WS_GWPART 16896
#define WS_GH     33280
#define WS_GW     34304
#define WS_ANOM   35328

// ---------------- K1: two 256-entry softmaxes ----------------
__global__ void k1_softmax(const float* __restrict__ dwh,
                           const float* __restrict__ dww,
                           float* __restrict__ ws) {
    __shared__ float red[256];
    int t = threadIdx.x;
    for (int pass = 0; pass < 2; ++pass) {
        const float* src = (pass == 0) ? dww : dwh;     // ww from dir_weight_w, wh from dir_weight_h
        float* dst = ws + ((pass == 0) ? WS_WW : WS_WH);
        float v = src[t];
        red[t] = v; __syncthreads();
        for (int s = 128; s > 0; s >>= 1) { if (t < s) red[t] = fmaxf(red[t], red[t + s]); __syncthreads(); }
        float m = red[0]; __syncthreads();
        float e = __expf(v - m);
        red[t] = e; __syncthreads();
        for (int s = 128; s > 0; s >>= 1) { if (t < s) red[t] += red[t + s]; __syncthreads(); }
        float tot = red[0]; __syncthreads();
        dst[t] = e / tot;
        __syncthreads();
    }
}

// ---------------- K2: weighted row/col sums, x read once via LDS plane ----------------
__global__ __launch_bounds__(256) void k2_partials(const float* __restrict__ x,
                                                   float* __restrict__ ws) {
    __shared__ float plane[64 * 65];
    __shared__ float pR[256];
    __shared__ float pC[256];
    __shared__ float accH[64];
    __shared__ float accW[64];
    int t = threadIdx.x;
    int b = blockIdx.x;      // 0..15
    int g = blockIdx.y;      // 0..15 channel group

    if (t < 64) { accH[t] = 0.f; accW[t] = 0.f; }
    __syncthreads();

    for (int ci = 0; ci < 16; ++ci) {
        int c = g * 16 + ci;
        const float* xp = x + ((b * Cn + c) * Hn) * Wn;
        for (int i = t; i < PLANE; i += 256)
            plane[(i >> 6) * 65 + (i & 63)] = xp[i];
        __syncthreads();

        int hh = t >> 2, q = t & 3;
        float s = 0.f, s2 = 0.f;
        #pragma unroll
        for (int k = 0; k < 16; ++k) {
            s  += plane[hh * 65 + q * 16 + k];          // row sums
            s2 += plane[(q * 16 + k) * 65 + hh];        // col sums
        }
        pR[t] = s; pC[t] = s2;
        __syncthreads();
        if (t < 64) {
            float rs = pR[t * 4] + pR[t * 4 + 1] + pR[t * 4 + 2] + pR[t * 4 + 3];
            float cs = pC[t * 4] + pC[t * 4 + 1] + pC[t * 4 + 2] + pC[t * 4 + 3];
            accH[t] += ws[WS_WW + c] * rs;
            accW[t] += ws[WS_WH + c] * cs;
        }
        __syncthreads();
    }
    if (t < 64) {
        ws[WS_GHPART + (b * 16 + g) * 64 + t] = accH[t];
        ws[WS_GWPART + (b * 16 + g) * 64 + t] = accW[t];
    }
}

// ---------------- K3: final GH/GW ----------------
__global__ void k3_gates(float* __restrict__ ws) {
    int b = blockIdx.x;
    int t = threadIdx.x;        // 128 threads: 0..63 -> GH, 64..127 -> GW
    const float scale = 1.f / (float)(Cn * Wn);   // C*W == C*H == 16384
    if (t < 64) {
        float s = 0.f;
        for (int g = 0; g < 16; ++g) s += ws[WS_GHPART + (b * 16 + g) * 64 + t];
        ws[WS_GH + b * 64 + t] = s * scale;
    } else {
        int w = t - 64;
        float s = 0.f;
        for (int g = 0; g < 16; ++g) s += ws[WS_GWPART + (b * 16 + g) * 64 + w];
        ws[WS_GW + b * 64 + w] = s * scale;
    }
}

// ---------------- K4: consistency via async-to-LDS staging + box filter + WMMA ----------------
// grid = (4 wtiles, 8 htiles, 16 b), 256 threads = 8 waves.
// Per 4-channel group: stage raw x1/x2 halo tiles (10x18 per channel) into LDS with
// GLOBAL_LOAD_ASYNC_TO_LDS_B32 (addresses pre-clamped in-bounds; zero-padding applied in
// the masked product pass), s_wait_asynccnt + barrier, compute prod in LDS, then each
// lane takes |3x3 box sum| for 2 channels of its pixel and one v_wmma_f32_16x16x4_f32
// per wave accumulates the channel sum: D = ones(16x4) x B(4ch x 16pix) + C.
#define REG (4 * 10 * 18)   // 720 staged elements per group
__global__ __launch_bounds__(256) void k4_consistency(const float* __restrict__ x,
                                                      float* __restrict__ ws) {
    __shared__ float sx1[REG];
    __shared__ float sx2[REG];
    __shared__ float pst[REG];
    int t = threadIdx.x;
    int wt = blockIdx.x;   // 0..3
    int ht = blockIdx.y;   // 0..7
    int b  = blockIdx.z;   // 0..15
    int h0 = ht * 8, w0 = wt * 16;

    int lane = t & 31;
    int wave = t >> 5;                 // tile row 0..7
    int pc   = lane & 15;              // pixel col in tile
    int jbase = (lane >> 4) * 2;       // channels jbase, jbase+1 of the staged 4

    // ---- precompute per-thread staging descriptors (loop-invariant) ----
    unsigned lds1[3], lds2[3];
    int      goff[3];                  // clamped global offset within batch, channel term excluded
    unsigned validmask = 0;
    #pragma unroll
    for (int i = 0; i < 3; ++i) {
        int e = t + 256 * i;
        int ec = (e < REG) ? e : (REG - 1);
        int j = ec / 180;
        int rem = ec - j * 180;
        int r = rem / 18;
        int col = rem - r * 18;
        int gh = h0 - 1 + r;
        int gw = w0 - 1 + col;
        bool valid = (e < REG) && (gh >= 0) && (gh < Hn) && (gw >= 0) && (gw < Wn);
        if (valid) validmask |= (1u << i);
        int ghc = gh < 0 ? 0 : (gh > Hn - 1 ? Hn - 1 : gh);
        int gwc = gw < 0 ? 0 : (gw > Wn - 1 ? Wn - 1 : gw);
        goff[i] = ((b * Cn + j) * Hn + ghc) * Wn + gwc;
        lds1[i] = (unsigned)(uintptr_t)&sx1[ec];
        lds2[i] = (unsigned)(uintptr_t)&sx2[ec];
    }

    v2f aones; aones.x = 1.f; aones.y = 1.f;
    v8f cacc = {0.f, 0.f, 0.f, 0.f, 0.f, 0.f, 0.f, 0.f};

    for (int g4 = 0; g4 < 32; ++g4) {
        int c0 = g4 * 4;
        // ---- async stage raw x1/x2 into LDS ----
        #pragma unroll
        for (int i = 0; i < 3; ++i) {
            if (t + 256 * i < REG) {
                unsigned long long a1 =
                    (unsigned long long)(uintptr_t)(x + goff[i] + c0 * PLANE);
                unsigned long long a2 =
                    (unsigned long long)(uintptr_t)(x + goff[i] + (c0 + HALF) * PLANE);
                asm volatile("global_load_async_to_lds_b32 %0, %1, off"
                             :: "v"(lds1[i]), "v"(a1) : "memory");
                asm volatile("global_load_async_to_lds_b32 %0, %1, off"
                             :: "v"(lds2[i]), "v"(a2) : "memory");
            }
        }
        asm volatile("s_wait_asynccnt 0x0" ::: "memory");
        __syncthreads();

        // ---- masked product pass: prod = x1*x2, zero outside the image ----
        #pragma unroll
        for (int i = 0; i < 3; ++i) {
            int e = t + 256 * i;
            if (e < REG) {
                float v = sx1[e] * sx2[e];
                pst[e] = ((validmask >> i) & 1u) ? v : 0.f;
            }
        }
        __syncthreads();

        // ---- two channels per lane: |3x3 box sum| at pixel (h0+wave, w0+pc) ----
        float vch[2];
        #pragma unroll
        for (int jj = 0; jj < 2; ++jj) {
            int j = jbase + jj;
            const float* p = &pst[j * 180];
            float s = 0.f;
            #pragma unroll
            for (int dr = 0; dr < 3; ++dr) {
                int ro = (wave + dr) * 18 + pc;
                s += p[ro] + p[ro + 1] + p[ro + 2];
            }
            vch[jj] = fabsf(s);
        }
        v2f bop; bop.x = vch[0]; bop.y = vch[1];
        cacc = __builtin_amdgcn_wmma_f32_16x16x4_f32(
            false, aones, false, bop, (short)0, cacc, false, false);
        __syncthreads();
    }

    // D row0 (and row8) in cacc[0]: per-lane value = column sum for pixel (lane&15)
    float sumabs = cacc[0];
    float consistency = sumabs * (1.f / (9.f * 128.f));
    float anom = 1.f - fminf(fmaxf(consistency, 0.f), 1.f);
    if (lane < 16) {
        int h = h0 + wave, w = w0 + pc;
        ws[WS_ANOM + (b * Hn + h) * Wn + w] = anom;
    }
}

// ---------------- K5: streaming apply (float4) ----------------
__global__ __launch_bounds__(256) void k5_apply(const float* __restrict__ x,
                                                const float* __restrict__ ws,
                                                const float* __restrict__ gwp,
                                                const float* __restrict__ gbp,
                                                float* __restrict__ out) {
    int tid = blockIdx.x * 256 + threadIdx.x;   // 4,194,304 threads
    int base = tid * 4;                          // 4 consecutive w in one row
    int plane = base >> 12;                      // (b*256 + c)
    int b = plane >> 8;
    int h = (base >> 6) & 63;
    int w = base & 63;
    float gw = *gwp, gb = *gbp;
    float gh_v = ws[WS_GH + b * 64 + h];
    const float* anomp = ws + WS_ANOM + (b * Hn + h) * Wn + w;
    const float* gwrow = ws + WS_GW + b * 64 + w;

    float4 xv = *reinterpret_cast<const float4*>(x + base);
    float4 ov;
    float f0 = (gh_v + gwrow[0]) * anomp[0];
    float f1 = (gh_v + gwrow[1]) * anomp[1];
    float f2 = (gh_v + gwrow[2]) * anomp[2];
    float f3 = (gh_v + gwrow[3]) * anomp[3];
    ov.x = xv.x / (1.f + __expf(-(gw * f0 + gb)));
    ov.y = xv.y / (1.f + __expf(-(gw * f1 + gb)));
    ov.z = xv.z / (1.f + __expf(-(gw * f2 + gb)));
    ov.w = xv.w / (1.f + __expf(-(gw * f3 + gb)));
    *reinterpret_cast<float4*>(out + base) = ov;
}

extern "C" void kernel_launch(void* const* d_in, const int* in_sizes, int n_in,
                              void* d_out, int out_size, void* d_ws, size_t ws_size,
                              hipStream_t stream) {
    const float* x   = (const float*)d_in[0];
    const float* dwh = (const float*)d_in[1];   // dir_weight_h -> wh (softmax) for global_w
    const float* dww = (const float*)d_in[2];   // dir_weight_w -> ww (softmax) for global_h
    const float* gwp = (const float*)d_in[3];
    const float* gbp = (const float*)d_in[4];
    float* out = (float*)d_out;
    float* ws  = (float*)d_ws;

    k1_softmax<<<1, 256, 0, stream>>>(dwh, dww, ws);
    k2_partials<<<dim3(16, 16), 256, 0, stream>>>(x, ws);
    k3_gates<<<16, 128, 0, stream>>>(ws);
    k4_consistency<<<dim3(4, 8, 16), 256, 0, stream>>>(x, ws);
    k5_apply<<<16384, 256, 0, stream>>>(x, ws, gwp, gbp, out);
}